// AttributeQuantizer_84928683311592
// MI455X (gfx1250) — compile-verified
//
#include <hip/hip_runtime.h>

typedef __attribute__((ext_vector_type(2))) float v2f;
typedef __attribute__((ext_vector_type(8))) float v8f;

#define VQ_N 262144
#define VQ_K 512
#define VQ_D 64

// ---------------------------------------------------------------------------
// Kernel A: L2-normalize the codebook (512 x 64) into workspace.
// One wave per row; also emits perplexity = 1.0.
// ---------------------------------------------------------------------------
__global__ void __launch_bounds__(32)
vq_norm_embed(const float* __restrict__ emb, float* __restrict__ en,
              float* __restrict__ out_perp) {
    const int row  = blockIdx.x;     // 0..511
    const int lane = threadIdx.x;    // 0..31
    float x0 = emb[row * VQ_D + lane];
    float x1 = emb[row * VQ_D + 32 + lane];
    float s = x0 * x0 + x1 * x1;
    #pragma unroll
    for (int m = 1; m < 32; m <<= 1) s += __shfl_xor(s, m, 32);
    const float inv = 1.0f / fmaxf(sqrtf(s), 1e-12f);
    en[row * VQ_D + lane]      = x0 * inv;
    en[row * VQ_D + 32 + lane] = x1 * inv;
    if (row == 0 && lane == 0) out_perp[0] = 1.0f;
}

// ---------------------------------------------------------------------------
// Kernel B: one wave per 16 input rows. fp32 WMMA 16x16x4 over D=64,
// looped across all 32 column tiles (K=512). Epilogue: argmax, one-hot,
// quantized gather, loss atomic.
// ---------------------------------------------------------------------------
__global__ void __launch_bounds__(32)
vq_main(const float* __restrict__ X, const int* __restrict__ labels,
        const float* __restrict__ emb, const float* __restrict__ en,
        float* __restrict__ out) {
    const int m0   = blockIdx.x * 16;
    const int lane = threadIdx.x;
    const int half = lane >> 4;       // 0 or 1
    const int lid  = lane & 15;

    __shared__ float lds_x[16][VQ_D];
    __shared__ float lds_scale[16];
    __shared__ int   lds_label[16];
    __shared__ int   lds_best[16];

    // Stage 16x64 input tile into LDS (32 floats per lane).
    #pragma unroll
    for (int i = lane; i < 16 * VQ_D; i += 32) {
        const int r = i >> 6, c = i & 63;
        lds_x[r][c] = X[(size_t)(m0 + r) * VQ_D + c];
    }
    if (lane < 16) lds_label[lane] = labels[m0 + lane];
    __syncthreads();

    if (lane < 16) {
        float s = 0.f;
        #pragma unroll
        for (int c = 0; c < VQ_D; ++c) { const float v = lds_x[lane][c]; s += v * v; }
        lds_scale[lane] = 1.0f / fmaxf(sqrtf(s), 1e-12f);
    }
    __syncthreads();

    // A fragments: 16 k-steps of (16x4). Same lane->K map used for A and B,
    // so the accumulated dot is layout-permutation invariant.
    float a0[16], a1[16];
    {
        const float sc = lds_scale[lid];
        #pragma unroll
        for (int ks = 0; ks < 16; ++ks) {
            const int k = ks * 4 + 2 * half;
            a0[ks] = lds_x[lid][k]     * sc;
            a1[ks] = lds_x[lid][k + 1] * sc;
        }
    }

    float bestVal[8], picked[8];
    int   bestIdx[8], myLab[8];
    #pragma unroll
    for (int r = 0; r < 8; ++r) {
        bestVal[r] = -3.0e38f; bestIdx[r] = 0x7fffffff; picked[r] = 0.f;
        myLab[r]   = lds_label[r + 8 * half];
    }

    for (int nt = 0; nt < VQ_K / 16; ++nt) {
        const float* eb = en + (size_t)(nt * 16 + lid) * VQ_D;
        v8f acc = {};
        #pragma unroll
        for (int ks = 0; ks < 16; ++ks) {
            const int k = ks * 4 + 2 * half;
            v2f A; A.x = a0[ks];  A.y = a1[ks];
            v2f B; B.x = eb[k];   B.y = eb[k + 1];
            acc = __builtin_amdgcn_wmma_f32_16x16x4_f32(
                false, A, false, B, (short)0, acc, false, false);
        }
        const int col = nt * 16 + lid;
        #pragma unroll
        for (int r = 0; r < 8; ++r) {
            const float v = acc[r];   // row = r + 8*half, column = col
            if (v > bestVal[r]) { bestVal[r] = v; bestIdx[r] = col; }
            if (col == myLab[r]) picked[r] = v;
        }
    }

    // Butterfly reduce across the 16 lanes of each half (wave32 shuffles).
    #pragma unroll
    for (int m = 1; m < 16; m <<= 1) {
        #pragma unroll
        for (int r = 0; r < 8; ++r) {
            const float ov = __shfl_xor(bestVal[r], m, 32);
            const int   oi = __shfl_xor(bestIdx[r], m, 32);
            if (ov > bestVal[r] || (ov == bestVal[r] && oi < bestIdx[r])) {
                bestVal[r] = ov; bestIdx[r] = oi;
            }
            picked[r] += __shfl_xor(picked[r], m, 32);  // exactly one lane nonzero
        }
    }

    if (lid == 0) {
        float s = 0.f;
        #pragma unroll
        for (int r = 0; r < 8; ++r) {
            lds_best[r + 8 * half] = bestIdx[r];
            s += (1.0f - picked[r]);
        }
        atomicAdd(&out[0], s * (1.0f / (float)VQ_N));
    }
    __syncthreads();

    const size_t off_q = 1;
    const size_t off_e = 2 + (size_t)VQ_N * VQ_D;          // after quantized+perplexity
    const size_t off_i = off_e + (size_t)VQ_N * VQ_K;      // indices tail

    if (lane < 16) {
        out[off_i + m0 + lane] = (float)lds_best[lane];
    }
    // Quantized: 16 rows x 64, gather from 128KB codebook (L2-resident).
    #pragma unroll
    for (int i = lane; i < 16 * VQ_D; i += 32) {
        const int r = i >> 6, c = i & 63;
        out[off_q + (size_t)(m0 + r) * VQ_D + c] =
            emb[(size_t)lds_best[r] * VQ_D + c];
    }
    // One-hot rows written in full as float4 (no separate 512MB memset pass).
    for (int i = lane; i < 16 * (VQ_K / 4); i += 32) {
        const int r = i >> 7, c4 = i & 127;
        const int idx = lds_best[r];
        float4 v = make_float4(0.f, 0.f, 0.f, 0.f);
        if ((idx >> 2) == c4) (&v.x)[idx & 3] = 1.0f;
        reinterpret_cast<float4*>(out + off_e + (size_t)(m0 + r) * VQ_K)[c4] = v;
    }
}

extern "C" void kernel_launch(void* const* d_in, const int* in_sizes, int n_in,
                              void* d_out, int out_size, void* d_ws, size_t ws_size,
                              hipStream_t stream) {
    (void)in_sizes; (void)n_in; (void)out_size; (void)ws_size;
    const float* X      = (const float*)d_in[0];
    const int*   labels = (const int*)d_in[1];
    const float* emb    = (const float*)d_in[2];
    float*       out    = (float*)d_out;
    float*       en     = (float*)d_ws;   // 512*64 floats = 128 KB normalized codebook

    // Loss is atomically accumulated -> must be zeroed every call (deterministic).
    hipMemsetAsync(d_out, 0, sizeof(float), stream);

    const size_t off_p = 1 + (size_t)VQ_N * VQ_D;   // perplexity slot
    vq_norm_embed<<<VQ_K, 32, 0, stream>>>(emb, en, out + off_p);
    vq_main<<<VQ_N / 16, 32, 0, stream>>>(X, labels, emb, en, out);
}